// DSBatchNorm2_38087769981102
// MI455X (gfx1250) — compile-verified
//
#include <hip/hip_runtime.h>
#include <hip/hip_bf16.h>

#define NF 1024     // features
#define NB 65536    // batch rows
#define NDOM 8
#define FB 128      // features per block (8 waves x 16)
#define TR 32       // tile rows staged in LDS per buffer
#define CHUNK 1024  // rows per block
#define NTILES (CHUNK / TR)

typedef float v2f __attribute__((ext_vector_type(2)));
typedef float v8f __attribute__((ext_vector_type(8)));
typedef int   vi4 __attribute__((ext_vector_type(4)));
typedef vi4 __attribute__((address_space(1)))* g4_t;   // global int4*
typedef vi4 __attribute__((address_space(3)))* l4_t;   // LDS int4*

// ---------------------------------------------------------------------------
// Async-copy helpers (GLOBAL_LOAD_ASYNC_TO_LDS_B128 + s_wait_asynccnt)
// ---------------------------------------------------------------------------
__device__ __forceinline__ void wait_async_le4() {
#if __has_builtin(__builtin_amdgcn_s_wait_asynccnt)
  __builtin_amdgcn_s_wait_asynccnt(4);
#else
  asm volatile("s_wait_asynccnt 0x4" ::: "memory");
#endif
}
__device__ __forceinline__ void wait_async_le0() {
#if __has_builtin(__builtin_amdgcn_s_wait_asynccnt)
  __builtin_amdgcn_s_wait_asynccnt(0);
#else
  asm volatile("s_wait_asynccnt 0x0" ::: "memory");
#endif
}

__device__ __forceinline__ void async_copy16(const float* gsrc, float* ldst,
                                             unsigned long long base,
                                             unsigned voff) {
#if __has_builtin(__builtin_amdgcn_global_load_async_to_lds_b128)
  (void)base; (void)voff;
  __builtin_amdgcn_global_load_async_to_lds_b128(
      (g4_t)(size_t)gsrc, (l4_t)ldst, 0, 0);
#else
  unsigned lds_off = (unsigned)(size_t)(l4_t)ldst;
  asm volatile("global_load_async_to_lds_b128 %0, %1, %2"
               :: "v"(lds_off), "v"(voff), "s"(base) : "memory");
#endif
}

// ---------------------------------------------------------------------------
// Kernel A: per-domain counts (LDS pre-reduction -> 16 global atomics/block)
// ---------------------------------------------------------------------------
__global__ void count_kernel(const long long* __restrict__ y,
                             float* __restrict__ cnt, int n) {
  __shared__ float lc[16];
  if (threadIdx.x < 16) lc[threadIdx.x] = 0.0f;
  __syncthreads();
  for (int i = blockIdx.x * blockDim.x + threadIdx.x; i < n;
       i += gridDim.x * blockDim.x) {
    atomicAdd(&lc[(int)y[i] & 15], 1.0f);
  }
  __syncthreads();
  if (threadIdx.x < 16) atomicAdd(&cnt[threadIdx.x], lc[threadIdx.x]);
}

// ---------------------------------------------------------------------------
// Kernel B: segmented sums via one-hot WMMA, x staged through LDS with
// double-buffered async b128 copies.
//   D[16dom,16feat] += onehot[16x4] * x[4x16]   (and x*x for sum-of-squares)
// A layout (f32 16x4): lane l -> M = l%16, K(vgpr v) = 2*(l/16) + v
// B layout (f32 4x16): lane l -> N = l%16, K(vgpr v) = 2*(l/16) + v
// D layout: vgpr v, lane half h -> M = v + 8h, N = l%16
// ---------------------------------------------------------------------------
__global__ void seg_sums_wmma(const float* __restrict__ x,
                              const long long* __restrict__ y,
                              float* __restrict__ gsum,
                              float* __restrict__ gss) {
  __shared__ float tile[2][TR * FB];   // 2 x 16KB

  const int lane = threadIdx.x & 31;
  const int wave = threadIdx.x >> 5;   // 0..7
  const int lh   = lane >> 4;          // lane half
  const int ln   = lane & 15;          // M (A) / N (B)
  const int fl   = wave * 16 + ln;     // feature within 128-block
  const int f0b  = blockIdx.x * FB;    // block feature base
  const int r0   = blockIdx.y * CHUNK; // block row base
  const unsigned long long xbase = (unsigned long long)(size_t)x;

  // Stage TR rows x FB feats: 256 threads x 4 rounds of async b128 (16B/lane).
  auto fill = [&](int buf, int rt) {
#pragma unroll
    for (int j = 0; j < TR / 8; ++j) {
      const int rl = j * 8 + wave;                 // 8 rows per round
      const size_t goff = ((size_t)(r0 + rt + rl) * NF + f0b + lane * 4);
      async_copy16(x + goff, &tile[buf][rl * FB + lane * 4], xbase,
                   (unsigned)(goff * sizeof(float)));
    }
  };

  v8f csum = {};
  v8f css  = {};

  fill(0, 0);
  for (int t = 0; t < NTILES; ++t) {
    const int buf = t & 1;
    if (t + 1 < NTILES) {
      fill(buf ^ 1, (t + 1) * TR);  // overlap next tile's fill
      wait_async_le4();             // in-order: current tile's 4 are done
    } else {
      wait_async_le0();
    }
    __syncthreads();

    const float* tb = tile[buf];
    const int rt = t * TR;
#pragma unroll
    for (int rl = 0; rl < TR; rl += 4) {
      const int gr = r0 + rt + rl + 2 * lh;
      const int ya = (int)y[gr];
      const int yb = (int)y[gr + 1];

      v2f a;
      a.x = (ya == ln) ? 1.0f : 0.0f;   // onehot[M=ln][K=2*lh]
      a.y = (yb == ln) ? 1.0f : 0.0f;   // onehot[M=ln][K=2*lh+1]

      v2f b;
      b.x = tb[(rl + 2 * lh) * FB + fl];
      b.y = tb[(rl + 2 * lh + 1) * FB + fl];
      v2f b2;
      b2.x = b.x * b.x;
      b2.y = b.y * b.y;

      csum = __builtin_amdgcn_wmma_f32_16x16x4_f32(
          false, a, false, b,  (short)0, csum, false, false);
      css  = __builtin_amdgcn_wmma_f32_16x16x4_f32(
          false, a, false, b2, (short)0, css,  false, false);
    }
    __syncthreads();
  }

  // Flush accumulators: lane (lh, ln), vgpr v -> domain m = v + 8*lh.
#pragma unroll
  for (int v = 0; v < 8; ++v) {
    const int m = v + 8 * lh;
    atomicAdd(&gsum[m * NF + f0b + fl], csum[v]);
    atomicAdd(&gss [m * NF + f0b + fl], css[v]);
  }
}

// ---------------------------------------------------------------------------
// Kernel S: mean / inv-std per (domain, feature).  var = E[x^2] - mean^2.
// ---------------------------------------------------------------------------
__global__ void stats_kernel(const float* __restrict__ gsum,
                             const float* __restrict__ gss,
                             const float* __restrict__ cnt,
                             float* __restrict__ mean,
                             float* __restrict__ invstd) {
  const int i = blockIdx.x * blockDim.x + threadIdx.x;  // 0..8191
  if (i >= NDOM * NF) return;
  const int d = i >> 10;
  const float c  = cnt[d];
  const float sc = fmaxf(c, 1.0f);
  const float m  = gsum[i] / sc;
  float var = gss[i] / sc - m * m;
  var = fmaxf(var, 0.0f);
  mean[i]   = m;
  invstd[i] = rsqrtf(var + 1e-5f);
}

// ---------------------------------------------------------------------------
// Kernel C: fused normalize.  One block per row (1024 feats / 256 thr = f4).
// ---------------------------------------------------------------------------
__global__ void norm_kernel(const float* __restrict__ x,
                            const long long* __restrict__ y,
                            const float* __restrict__ gamma,
                            const float* __restrict__ beta,
                            const float* __restrict__ mean,
                            const float* __restrict__ invstd,
                            const float* __restrict__ cnt,
                            float* __restrict__ out) {
  const int row = blockIdx.x;
  const int d   = (int)y[row];
  const float c = cnt[d];
  const int f   = threadIdx.x * 4;
  const size_t base = (size_t)row * NF + f;

  const float4 xv = *(const float4*)(x + base);
  float4 o;
  if (c > 1.0f) {
    const float4 mv = *(const float4*)(mean   + d * NF + f);
    const float4 sv = *(const float4*)(invstd + d * NF + f);
    const float4 gv = *(const float4*)(gamma + f);
    const float4 bv = *(const float4*)(beta  + f);
    o.x = gv.x * ((xv.x - mv.x) * sv.x) + bv.x;
    o.y = gv.y * ((xv.y - mv.y) * sv.y) + bv.y;
    o.z = gv.z * ((xv.z - mv.z) * sv.z) + bv.z;
    o.w = gv.w * ((xv.w - mv.w) * sv.w) + bv.w;
  } else if (c == 1.0f) {
    o = xv;
  } else {
    o.x = o.y = o.z = o.w = 0.0f;
  }
  *(float4*)(out + base) = o;
}

// ---------------------------------------------------------------------------
// Launch
// ---------------------------------------------------------------------------
extern "C" void kernel_launch(void* const* d_in, const int* in_sizes, int n_in,
                              void* d_out, int out_size, void* d_ws,
                              size_t ws_size, hipStream_t stream) {
  (void)in_sizes; (void)n_in; (void)out_size; (void)ws_size;
  const float*     x     = (const float*)d_in[0];
  const long long* y     = (const long long*)d_in[1];
  const float*     gamma = (const float*)d_in[2];
  const float*     beta  = (const float*)d_in[3];
  float* out = (float*)d_out;

  float* ws     = (float*)d_ws;
  float* gsum   = ws;                       // 16*1024
  float* gss    = ws + 16 * NF;             // 16*1024
  float* cnt    = ws + 32 * NF;             // 16
  float* mean   = ws + 32 * NF + 16;        // 8*1024 (16B aligned)
  float* invstd = mean + NDOM * NF;         // 8*1024

  // Zero accumulators every call (ws is poisoned once, never re-zeroed).
  (void)hipMemsetAsync(d_ws, 0, (32 * NF + 16) * sizeof(float), stream);

  count_kernel<<<256, 256, 0, stream>>>(y, cnt, NB);

  // 8 feature groups x 64 row chunks; 8 waves/block, 16 feats/wave.
  dim3 gB(NF / FB, NB / CHUNK);
  seg_sums_wmma<<<gB, 256, 0, stream>>>(x, y, gsum, gss);

  stats_kernel<<<(NDOM * NF + 255) / 256, 256, 0, stream>>>(
      gsum, gss, cnt, mean, invstd);

  norm_kernel<<<NB, 256, 0, stream>>>(x, y, gamma, beta, mean, invstd, cnt,
                                      out);
}